// GCN_35562329210944
// MI455X (gfx1250) — compile-verified
//
#include <hip/hip_runtime.h>

// GCN forward for MI455X (gfx1250), wave32.
// Pipeline: deg -> rdeg -> scatter-sum(F=128) -> WMMA GEMM1(+bias,ReLU)
//           -> scatter-sum(F=256) -> WMMA GEMM2(+bias) -> d_out
// GEMMs use V_WMMA_F32_16X16X4_F32 (fp32-exact; GEMM flops negligible vs the
// L2-bound edge aggregation). Each wave computes NT 16x16 tiles, reusing one
// A fragment across NT back-to-back WMMAs.

#define NN 50000     // nodes
#define NE 1600000   // edges
#define F0 128       // input feats
#define F1 256       // hidden
#define NC 41        // classes

typedef __attribute__((ext_vector_type(2))) float v2f;
typedef __attribute__((ext_vector_type(8))) float v8f;

__global__ void zero_f32(float* __restrict__ p, int n) {
  int i = blockIdx.x * blockDim.x + threadIdx.x;
  if (i < n) p[i] = 0.0f;
}

__global__ void deg_count(const int* __restrict__ dst, float* __restrict__ deg) {
  int e = blockIdx.x * blockDim.x + threadIdx.x;
  if (e < NE) atomicAdd(&deg[dst[e]], 1.0f);
}

__global__ void make_rdeg(float* __restrict__ deg) {
  int i = blockIdx.x * blockDim.x + threadIdx.x;
  if (i < NN) deg[i] = 1.0f / fmaxf(deg[i], 1.0f);
}

// One thread per (edge, group of 4 features). F = 4 << LOGG.
template <int LOGG>
__global__ void edge_scatter(const float* __restrict__ h,
                             const int* __restrict__ src,
                             const int* __restrict__ dst,
                             float* __restrict__ out) {
  const int F = 4 << LOGG;
  unsigned gid = blockIdx.x * blockDim.x + threadIdx.x;
  unsigned e = gid >> LOGG;
  if (e >= NE) return;
  int c = (int)(gid & ((1u << LOGG) - 1u)) << 2;
  int s = src[e];
  int d = dst[e];
  const float4 v = *(const float4*)(h + (size_t)s * F + c);
  float* o = out + (size_t)d * F + c;
  atomicAdd(o + 0, v.x);
  atomicAdd(o + 1, v.y);
  atomicAdd(o + 2, v.z);
  atomicAdd(o + 3, v.w);
}

// C[M,NCOLS] = act( (A .* rdeg[m]) @ W[K,NCOLS] + bias ).
// One wave computes NT adjacent 16x16 output tiles (16 rows x NT*16 cols),
// reusing a single A fragment across NT WMMAs per K-step.
// GUARD (NCOLS not multiple of 16) masks the padded columns of the last tile.
// All early exits are wave-uniform -> EXEC all-1s at every WMMA.
template <int K, int NCOLS, int NT, bool RELU>
__global__ void wmma_gemm(const float* __restrict__ A,
                          const float* __restrict__ rdeg,
                          const float* __restrict__ W,
                          const float* __restrict__ bias,
                          float* __restrict__ C,
                          int total_waves) {
  constexpr bool GUARD = (NCOLS % 16) != 0;
  constexpr int NTC = (NCOLS + 15) / 16;       // total col tiles
  constexpr int GROUPS = (NTC + NT - 1) / NT;  // tile-groups per row block

  int wave = threadIdx.x >> 5;
  int lane = threadIdx.x & 31;
  int w = blockIdx.x * (blockDim.x >> 5) + wave;
  if (w >= total_waves) return;                // wave-uniform

  int mtile = w / GROUPS;                      // GROUPS is 4 or 1 -> cheap
  int g = w - mtile * GROUPS;
  int m0 = mtile << 4;
  int n0 = g * (NT * 16);
  int half = lane >> 4;                        // 0: K pair {0,1}; 1: {2,3}
  int l16 = lane & 15;
  int mA = m0 + l16;

  float scale = rdeg[mA];
  const float* arow = A + (size_t)mA * K;

  int nB[NT];
  bool nv[NT];
  const float* wcol[NT];
#pragma unroll
  for (int t = 0; t < NT; ++t) {
    nB[t] = n0 + t * 16 + l16;
    nv[t] = !GUARD || (nB[t] < NCOLS);
    wcol[t] = W + (nv[t] ? nB[t] : 0);         // clamp: no OOB reads
  }

  v8f acc[NT];
#pragma unroll
  for (int t = 0; t < NT; ++t) acc[t] = (v8f){};

  for (int kb = 0; kb < K; kb += 4) {
    int k0 = kb + half * 2;
    float2 av = *(const float2*)(arow + k0);   // k0 even -> 8B aligned
    v2f a;
    a[0] = av.x * scale;
    a[1] = av.y * scale;
#pragma unroll
    for (int t = 0; t < NT; ++t) {
      v2f b;
      b[0] = wcol[t][(size_t)k0 * NCOLS];
      b[1] = wcol[t][(size_t)(k0 + 1) * NCOLS];
      if (GUARD && !nv[t]) { b[0] = 0.0f; b[1] = 0.0f; }
      // (neg_a, A, neg_b, B, c_mod, C, reuse_a, reuse_b)
      acc[t] = __builtin_amdgcn_wmma_f32_16x16x4_f32(
          false, a, false, b, (short)0, acc[t], false, false);
    }
  }

#pragma unroll
  for (int t = 0; t < NT; ++t) {
    float bv = nv[t] ? bias[nB[t]] : 0.0f;
#pragma unroll
    for (int r = 0; r < 8; ++r) {
      int m = m0 + r + half * 8;               // C layout: VGPR r -> row r/r+8
      float vout = acc[t][r] + bv;
      if (RELU) vout = fmaxf(vout, 0.0f);
      if (nv[t]) C[(size_t)m * NCOLS + nB[t]] = vout;
    }
  }
}

static inline unsigned nblk(long long n, int b) {
  return (unsigned)((n + b - 1) / b);
}

extern "C" void kernel_launch(void* const* d_in, const int* in_sizes, int n_in,
                              void* d_out, int out_size, void* d_ws, size_t ws_size,
                              hipStream_t stream) {
  const float* features = (const float*)d_in[0];
  const int*   src      = (const int*)d_in[1];
  const int*   dst      = (const int*)d_in[2];
  const float* W1       = (const float*)d_in[3];
  const float* b1       = (const float*)d_in[4];
  const float* W2       = (const float*)d_in[5];
  const float* b2       = (const float*)d_in[6];
  float* out = (float*)d_out;

  // Workspace layout (floats): rdeg[NN] | sum0[NN*F0] | h1[NN*F1] | sum1[NN*F1]
  float* ws   = (float*)d_ws;
  float* rdeg = ws;
  float* sum0 = ws + 50048;                 // 8B-aligned, past rdeg
  float* h1   = sum0 + (size_t)NN * F0;
  float* sum1 = h1 + (size_t)NN * F1;

  // Zero accumulators (d_ws is poisoned; must re-zero every call).
  zero_f32<<<nblk(NN, 256), 256, 0, stream>>>(rdeg, NN);
  zero_f32<<<nblk((long long)NN * F0, 256), 256, 0, stream>>>(sum0, NN * F0);
  zero_f32<<<nblk((long long)NN * F1, 256), 256, 0, stream>>>(sum1, NN * F1);

  // Degrees -> reciprocal degrees.
  deg_count<<<nblk(NE, 256), 256, 0, stream>>>(dst, rdeg);
  make_rdeg<<<nblk(NN, 256), 256, 0, stream>>>(rdeg);

  // Layer 0 aggregation: sum0[dst] += features[src]  (F=128 -> 32 groups/edge)
  edge_scatter<5><<<nblk((long long)NE * 32, 256), 256, 0, stream>>>(
      features, src, dst, sum0);

  // GEMM1: h1 = relu((sum0 * rdeg) @ W1 + b1)
  // 3125 row blocks x 4 groups of 4 tiles = 12500 waves
  {
    int tw = (NN / 16) * 4;
    wmma_gemm<F0, F1, 4, true><<<nblk(tw, 8), 256, 0, stream>>>(
        sum0, rdeg, W1, b1, h1, tw);
  }

  // Layer 1 aggregation: sum1[dst] += h1[src]  (F=256 -> 64 groups/edge)
  edge_scatter<6><<<nblk((long long)NE * 64, 256), 256, 0, stream>>>(
      h1, src, dst, sum1);

  // GEMM2: out = (sum1 * rdeg) @ W2 + b2, N=41 -> 3 tiles per wave (whole row)
  {
    int tw = NN / 16;                       // 3125 waves
    wmma_gemm<F1, NC, 3, false><<<nblk(tw, 8), 256, 0, stream>>>(
        sum1, rdeg, W2, b2, out, tw);
  }
}